// Splat_60662118088796
// MI455X (gfx1250) — compile-verified
//
#include <hip/hip_runtime.h>
#include <cstdint>

// Problem constants (from reference): B=4, H=4, V=8, N=16384, fd=32, NC=32^3
#define BATCH   4
#define HEADS   4
#define FD      32
#define VDIM    8
#define NPTS    16384
#define NCELLS  (32 * 32 * 32)
#define TILE_N  256

typedef unsigned int u32x4 __attribute__((ext_vector_type(4)));
typedef int          i32x8 __attribute__((ext_vector_type(8)));
typedef int          i32x4 __attribute__((ext_vector_type(4)));

// ---------------------------------------------------------------------------
// Kernel 1: zero-fill the 64 MB output (scatter-max starts from zeros).
// ---------------------------------------------------------------------------
__global__ __launch_bounds__(256) void splat_zero_kernel(float4* __restrict__ out) {
    size_t i = (size_t)blockIdx.x * blockDim.x + threadIdx.x;
    out[i] = make_float4(0.f, 0.f, 0.f, 0.f);
}

// ---------------------------------------------------------------------------
// Kernel 2: TDM-staged scatter-max.
//   block = (n-tile of 256, bh slice). Wave 0 DMAs the [FD x TILE_N] feature
//   tile (row stride NPTS) into LDS via TENSOR_LOAD_TO_LDS; everyone waits on
//   TENSORcnt + workgroup barrier; then per-thread register-resident coords/
//   indices drive global_atomic_max_i32 into the output grid.
// ---------------------------------------------------------------------------
__global__ __launch_bounds__(TILE_N) void splat_scatter_max_kernel(
    const float*     __restrict__ coord,     // [B,H,V,N] f32
    const long long* __restrict__ fidx,      // [B,H,V,N] i64, values < NCELLS
    const float*     __restrict__ features,  // [B,128,N] f32
    float*           __restrict__ out)       // [B,128,NCELLS] f32 (zeroed)
{
    __shared__ float smem[FD * TILE_N];      // 32 KB feature tile, [f][n]

    const int tile = blockIdx.x;             // 0..63   (n tile)
    const int bh   = blockIdx.y;             // 0..15   (b*HEADS + h)
    const int n0   = tile * TILE_N;
    const int t    = threadIdx.x;

    // ---- Tensor Data Mover: features[b, h*FD .. h*FD+31, n0 .. n0+255] -> LDS
    if ((threadIdx.x >> 5) == 0) {           // one wave issues the DMA
        const uint64_t gaddr =
            (uint64_t)(uintptr_t)(features + ((size_t)bh * FD) * NPTS + n0);
        const uint32_t lds_addr = (uint32_t)(uintptr_t)smem;  // low 32b = LDS offset

        // D# group 0: count=1 (valid), lds_addr, 57b global_addr, type=2
        u32x4 g0;
        g0.x = 1u;
        g0.y = lds_addr;
        g0.z = (uint32_t)(gaddr & 0xFFFFFFFFu);
        g0.w = (uint32_t)((gaddr >> 32) & 0x01FFFFFFu) | (2u << 30);

        // D# group 1: data_size=4B; tensor 256x32 (== tile, fully in-bounds);
        // tile_dim0=256 (contiguous n), tile_dim1=32 (f rows), stride0=NPTS.
        i32x8 g1;
        g1[0] = (int)(2u << 16);                              // data_size=2 (4B), mask=0
        g1[1] = (int)(((uint32_t)TILE_N & 0xFFFFu) << 16);    // tensor_dim0[15:0]
        g1[2] = (int)(((uint32_t)TILE_N >> 16) |
                      (((uint32_t)FD & 0xFFFFu) << 16));      // dim0[31:16] | dim1[15:0]
        g1[3] = (int)(((uint32_t)FD >> 16) |
                      (((uint32_t)TILE_N & 0xFFFFu) << 16));  // dim1[31:16] | tile_dim0
        g1[4] = (int)((uint32_t)FD & 0xFFFFu);                // tile_dim1 | tile_dim2=0
        g1[5] = (int)NPTS;                                    // tensor_dim0_stride[31:0]
        g1[6] = 0;                                            // stride0[47:32]|stride1 lo
        g1[7] = 0;                                            // stride1 hi

        i32x4 gz4 = {0, 0, 0, 0};                             // groups 2/3 unused (2-D)
        i32x8 gz8 = {0, 0, 0, 0, 0, 0, 0, 0};                 // extra group (clang-23 form)
        __builtin_amdgcn_tensor_load_to_lds(g0, g1, gz4, gz4, gz8, 0);
    }
    __builtin_amdgcn_s_wait_tensorcnt(0);    // issuing wave drains TENSORcnt
    __syncthreads();                         // tile visible to all 8 waves

    // ---- Per-thread n: preload 8 coords + 8 indices (coalesced B32/B64 loads)
    const size_t inBase = ((size_t)bh * VDIM) * NPTS + (size_t)(n0 + t);
    float cv[VDIM];
    int   ci[VDIM];
#pragma unroll
    for (int v = 0; v < VDIM; ++v) {
        const size_t p = inBase + (size_t)v * NPTS;
        cv[v] = coord[p];
        ci[v] = (int)fidx[p];                // indices < 32768, low word exact
    }

    int* __restrict__ iout = (int*)out;
    const size_t outBase0 = (size_t)bh * FD * (size_t)NCELLS;

#pragma unroll 4
    for (int f = 0; f < FD; ++f) {
        const float fv = smem[f * TILE_N + t];
        // coord >= 0 and out is zero-init: only fv > 0 can ever beat the init.
        if (fv > 0.0f) {
            const size_t ob = outBase0 + (size_t)f * NCELLS;
#pragma unroll
            for (int v = 0; v < VDIM; ++v) {
                const float val = fv * cv[v];          // val >= 0
                // positive IEEE floats are monotone as int bits -> exact max
                atomicMax(iout + ob + (size_t)ci[v], __float_as_int(val));
            }
        }
    }
}

// ---------------------------------------------------------------------------
extern "C" void kernel_launch(void* const* d_in, const int* in_sizes, int n_in,
                              void* d_out, int out_size, void* d_ws, size_t ws_size,
                              hipStream_t stream) {
    (void)in_sizes; (void)n_in; (void)d_ws; (void)ws_size;

    const float*     coord = (const float*)d_in[0];      // [4,4,8,16384] f32
    const long long* fidx  = (const long long*)d_in[1];  // [4,4,8,16384] i64
    const float*     feats = (const float*)d_in[2];      // [4,128,16384] f32
    float*           out   = (float*)d_out;              // [4,128,32768] f32

    // 16,777,216 floats -> 4,194,304 float4 stores
    const int zeroBlocks = (BATCH * 128 * NCELLS) / 4 / 256;
    splat_zero_kernel<<<zeroBlocks, 256, 0, stream>>>((float4*)out);

    dim3 grid(NPTS / TILE_N, BATCH * HEADS);             // (64, 16)
    splat_scatter_max_kernel<<<grid, TILE_N, 0, stream>>>(coord, fidx, feats, out);
}